// etofts_model_24936580120830
// MI455X (gfx1250) — compile-verified
//
#include <hip/hip_runtime.h>
#include <cstdint>

typedef float v4f __attribute__((ext_vector_type(4)));

#define N_TIME   60
#define HW_      65536                       // 256*256
#define DT_      5.0f
#define LOG2E_   1.44269504088896340736f
#define COSFA_   0.98480775301220805936f     // cos(10 deg)
#define QS_      (0.0225f * LOG2E_)          // R1*TR/1000 * log2(e)
#define EPS_     1e-8f

// One thread = 4 horizontally-adjacent pixels, full 60-step time scan in regs.
// 256 threads/block -> 1024 pixels/block -> 64 blocks per batch image.
__global__ __launch_bounds__(256) void etofts_kernel(
    const float* __restrict__ params,   // [B,3,H,W]
    const float* __restrict__ Cp,       // [B,60]
    const float* __restrict__ S0,       // [B,1,H,W]
    const float* __restrict__ T1,       // [B,1,H,W]
    float* __restrict__ out)            // [B,60,H,W]
{
    __shared__ float sCp[N_TIME];

    const int b      = blockIdx.x >> 6;   // 64 blocks per batch
    const int blkInB = blockIdx.x & 63;

    // ---- gfx1250 async global->LDS DMA: stage Cp[b, 0..59] once per block ----
    if (threadIdx.x < N_TIME) {
        uint32_t lds  = (uint32_t)(uintptr_t)&sCp[threadIdx.x]; // flat low 32b == LDS offset
        uint32_t voff = threadIdx.x * 4u;
        uint64_t base = (uint64_t)(uintptr_t)(Cp + b * N_TIME);
        asm volatile("global_load_async_to_lds_b32 %0, %1, %2"
                     :: "v"(lds), "v"(voff), "s"(base)
                     : "memory");
    }
    asm volatile("s_wait_asynccnt 0" ::: "memory");
    __syncthreads();

    const int pix = (blkInB * 256 + (int)threadIdx.x) * 4;

    const size_t pbase = (size_t)b * 3 * HW_ + pix;
    v4f kt4 = __builtin_nontemporal_load((const v4f*)(params + pbase));
    v4f vp4 = __builtin_nontemporal_load((const v4f*)(params + pbase + HW_));
    v4f ve4 = __builtin_nontemporal_load((const v4f*)(params + pbase + 2 * HW_));
    v4f s04 = __builtin_nontemporal_load((const v4f*)(S0 + (size_t)b * HW_ + pix));
    v4f t14 = __builtin_nontemporal_load((const v4f*)(T1 + (size_t)b * HW_ + pix));

    // Per-pixel loop-invariant state (8 regs x 4 pixels)
    float decay[4], Kq[4], vq[4], S0A[4], S0AeP[4], mcBeP[4], Beps[4], ce[4];
#pragma unroll
    for (int i = 0; i < 4; ++i) {
        float Ktrans = kt4[i] * (1.0f / 60.0f);
        float Kep    = Ktrans * __builtin_amdgcn_rcpf(ve4[i] + EPS_);
        decay[i]     = __builtin_amdgcn_exp2f(-Kep * (DT_ * LOG2E_));
        float P      = 5.0f * __builtin_amdgcn_rcpf(t14[i] + EPS_);
        float eP     = __builtin_amdgcn_exp2f(-P * LOG2E_);
        float B      = 1.0f - eP;
        S0A[i]       = s04[i] * (1.0f - COSFA_ * eP);   // S0*(1-cos*eP)
        S0AeP[i]     = S0A[i] * eP;
        mcBeP[i]     = -COSFA_ * B * eP;
        Beps[i]      = B + EPS_;
        Kq[i]        = QS_ * Ktrans;
        vq[i]        = QS_ * vp4[i];
        ce[i]        = 0.0f;
    }

    float* outp = out + (size_t)b * N_TIME * HW_ + pix;
#pragma unroll 4
    for (int t = 0; t < N_TIME; ++t) {
        const float cp   = sCp[t];
        const float cpDT = cp * DT_;
        v4f o;
#pragma unroll
        for (int i = 0; i < 4; ++i) {
            ce[i]      = fmaf(ce[i], decay[i], cpDT);          // Ce recurrence
            float earg = -fmaf(Kq[i], ce[i], vq[i] * cp);      // -log2e*R1*TR/1000*Ct
            float eQ   = __builtin_amdgcn_exp2f(earg);         // exp(-Q)
            float num  = fmaf(-S0AeP[i], eQ, S0A[i]);          // S0A*(1-eP*eQ)
            float den  = fmaf(mcBeP[i], eQ, Beps[i]);          // (1-cos*eP*eQ)*B+eps
            o[i]       = num * __builtin_amdgcn_rcpf(den);
        }
        // output written once, never reread: bypass L2 retention with NT store
        __builtin_nontemporal_store(o, (v4f*)(outp + (size_t)t * HW_));
    }
}

extern "C" void kernel_launch(void* const* d_in, const int* in_sizes, int n_in,
                              void* d_out, int out_size, void* d_ws, size_t ws_size,
                              hipStream_t stream) {
    const float* params = (const float*)d_in[0];
    const float* Cp     = (const float*)d_in[1];
    const float* S0     = (const float*)d_in[2];
    const float* T1     = (const float*)d_in[3];
    float* out          = (float*)d_out;

    const int B = in_sizes[1] / N_TIME;   // 8
    dim3 grid(B * 64), block(256);
    hipLaunchKernelGGL(etofts_kernel, grid, block, 0, stream,
                       params, Cp, S0, T1, out);
}